// BlockPatchMasking_72241349919073
// MI455X (gfx1250) — compile-verified
//
#include <hip/hip_runtime.h>
#include <stdint.h>

// Problem constants (from reference: B=64, P=16384, F=3, MULTI_MASK=2)
#define BATCH        64
#define MULTI        2
#define BM           (BATCH * MULTI)   // 128 rows
#define NPTS         16384
#define NDIM         3
#define NUM_MASKS    9830              // round(0.6 * P)
#define BLK_K        819              // int(0.05 * P)
#define NCEN         10               // round(P * (0.6-0.1)/819)
#define NTHREADS     256
#define NWAVES       (NTHREADS / 32)
#define NTILES       (NPTS / 16)      // 1024 WMMA column tiles per row
#define COORD_BYTES  (NPTS * NDIM * 4)  // 196608 B staged in LDS (CDNA5: 320KB/WG)

typedef float v2f __attribute__((ext_vector_type(2)));
typedef float v8f __attribute__((ext_vector_type(8)));

// Map float to monotonically ordered unsigned bits (handles negatives from FP rounding).
__device__ __forceinline__ uint32_t ford(float f) {
    uint32_t u = __float_as_uint(f);
    return u ^ ((u & 0x80000000u) ? 0xFFFFFFFFu : 0x80000000u);
}

// One 16(center)x16(point) distance tile via V_WMMA_F32_16X16X4_F32.
// A row m = (-2*sx, -2*sy, -2*sz, |s|^2), B col n = (px, py, pz, 1)
// => acc[c][p] = -2 s.p + |s|^2 ; caller adds |p|^2.
// A layout (32-bit 16x4): lanes 0-15 hold K=0,1 in v0,v1; lanes 16-31 hold K=2,3.
// B layout (4x16): lanes 0-15 col N=lane K=0,1; lanes 16-31 col N=lane-16 K=2,3.
// Coordinates are read from the LDS-staged copy (ds_load path).
__device__ __forceinline__ void tile_dist(const float* __restrict__ sco,
                                          int pbase, int lmod, int hi,
                                          v2f areg, v8f& acc, float& pnorm) {
    const int p = pbase + lmod;
    const float* pp = sco + (size_t)p * NDIM;
    float px = pp[0], py = pp[1], pz = pp[2];
    v2f breg;
    breg.x = hi ? pz : px;
    breg.y = hi ? 1.0f : py;
    v8f c0 = {};
    acc = __builtin_amdgcn_wmma_f32_16x16x4_f32(
        /*neg_a=*/false, areg, /*neg_b=*/false, breg,
        /*c_mod=*/(short)0, c0, /*reuse_a=*/false, /*reuse_b=*/false);
    pnorm = px * px + py * py + pz * pz;
}

__global__ __launch_bounds__(NTHREADS)
void BlockPatchMasking_72241349919073_kernel(const float* __restrict__ centers,
                                             const float* __restrict__ rand_centers,
                                             const float* __restrict__ rand_mask,
                                             float* __restrict__ out) {
    const int bm   = blockIdx.x;          // masked-row index [0, 128)
    const int tid  = threadIdx.x;
    const int lane = tid & 31;
    const int wave = tid >> 5;
    const int hi   = lane >> 4;           // which K-half / M-half this lane covers
    const int lmod = lane & 15;

    extern __shared__ float sco[];            // [NPTS*3] staged coordinates (192KB)

    __shared__ uint32_t hist[NCEN * 256];     // radix histograms (reused)
    __shared__ uint32_t bits[NPTS / 32];      // block-union bitmask
    __shared__ float    As[16][4];            // padded A matrix rows
    __shared__ uint64_t red[NTHREADS];        // reduction scratch
    __shared__ int      chosen[NCEN];
    __shared__ uint64_t prefix[NCEN];
    __shared__ uint32_t kth[NCEN];
    __shared__ uint64_t thresh2;
    __shared__ uint32_t remaining_s;

    const float* cen = centers + (size_t)(bm >> 1) * NPTS * NDIM;  // jnp.repeat axis0
    const float* rc  = rand_centers + (size_t)bm * NPTS;
    const float* rm  = rand_mask    + (size_t)bm * NPTS;

    // ---- Async-stage this row's coordinates global -> LDS (ASYNCcnt path) ----
    // 196608 B as 12288 x b128; overlapped with Phase 0 which only reads rand_centers.
    {
        const uint64_t gbase = (uint64_t)(uintptr_t)cen;
        const uint32_t lbase = (uint32_t)(uintptr_t)sco;   // low 32b of generic = LDS offset
        for (int i = tid; i < COORD_BYTES / 16; i += NTHREADS) {
            uint32_t la = lbase + (uint32_t)i * 16u;
            uint64_t ga = gbase + (uint64_t)i * 16u;
            asm volatile("global_load_async_to_lds_b128 %0, %1, off"
                         :: "v"(la), "v"(ga)
                         : "memory");
        }
    }

    // ---- Phase 0: 10 smallest rand_centers (argsort[:10]) via iterated argmin ----
    for (int t = 0; t < NCEN; ++t) {
        uint64_t best = ~0ull;
        for (int p = tid; p < NPTS; p += NTHREADS) {
            if (p + NTHREADS < NPTS)
                __builtin_prefetch(rc + p + NTHREADS, 0, 1);
            bool skip = false;
            for (int j = 0; j < t; ++j)
                if (chosen[j] == p) skip = true;
            if (!skip) {
                uint64_t key = ((uint64_t)ford(rc[p]) << 14) | (uint32_t)p;
                if (key < best) best = key;
            }
        }
        red[tid] = best;
        __syncthreads();
        for (int s = NTHREADS / 2; s > 0; s >>= 1) {
            if (tid < s && red[tid + s] < red[tid]) red[tid] = red[tid + s];
            __syncthreads();
        }
        if (tid == 0) chosen[t] = (int)(red[0] & 0x3FFFull);
        __syncthreads();
    }

    // ---- Drain this wave's async copies, then make LDS coords visible to all ----
    asm volatile("s_wait_asynccnt 0x0" ::: "memory");
    __syncthreads();

    // ---- Build A matrix rows (-2s, |s|^2) from LDS; pad centers 10..15 with zeros ----
    if (tid < 16) {
        float x = 0.f, y = 0.f, z = 0.f;
        if (tid < NCEN) {
            const float* sp = sco + (size_t)chosen[tid] * NDIM;
            x = sp[0]; y = sp[1]; z = sp[2];
        }
        As[tid][0] = -2.f * x;
        As[tid][1] = -2.f * y;
        As[tid][2] = -2.f * z;
        As[tid][3] = x * x + y * y + z * z;
    }
    for (int i = tid; i < NCEN * 256; i += NTHREADS) hist[i] = 0;
    if (tid < NCEN) { prefix[tid] = 0; kth[tid] = BLK_K; }
    __syncthreads();

    v2f areg;
    areg.x = As[lmod][hi * 2 + 0];
    areg.y = As[lmod][hi * 2 + 1];

    // ---- Phase 1: per-center radix-select of the 819th-smallest 46-bit key ----
    // key = ford(d2)[32b] << 14 | point_index[14b]  (stable tie-break like argsort)
    for (int pass = 0; pass < 6; ++pass) {
        const int dsh = 40 - 8 * pass;
        for (int tile = wave; tile < NTILES; tile += NWAVES) {
            const int pbase = tile * 16;
            v8f acc; float pnorm;
            tile_dist(sco, pbase, lmod, hi, areg, acc, pnorm);
            const uint32_t p = (uint32_t)(pbase + lmod);
#pragma unroll
            for (int j = 0; j < 8; ++j) {
                const int c = j + 8 * hi;
                if (c < NCEN) {
                    float d2 = acc[j] + pnorm;
                    uint64_t key = ((uint64_t)ford(d2) << 14) | p;
                    if ((key >> (dsh + 8)) == prefix[c])
                        atomicAdd(&hist[c * 256 + ((uint32_t)(key >> dsh) & 0xFFu)], 1u);
                }
            }
        }
        __syncthreads();
        if (tid < NCEN) {
            uint32_t k = kth[tid], cum = 0;
            int d = 0;
            for (; d < 256; ++d) {
                uint32_t h = hist[tid * 256 + d];
                if (cum + h >= k) break;
                cum += h;
            }
            if (d > 255) d = 255;
            prefix[tid] = (prefix[tid] << 8) | (uint32_t)d;
            kth[tid] = k - cum;
        }
        __syncthreads();
        for (int i = tid; i < NCEN * 256; i += NTHREADS) hist[i] = 0;
        __syncthreads();
    }

    // ---- Phase 2: block-union membership (key <= kth key => rank <= 819) ----
    for (int i = tid; i < NPTS / 32; i += NTHREADS) bits[i] = 0;
    __syncthreads();
    for (int tile = wave; tile < NTILES; tile += NWAVES) {
        const int pbase = tile * 16;
        v8f acc; float pnorm;
        tile_dist(sco, pbase, lmod, hi, areg, acc, pnorm);
        const uint32_t p = (uint32_t)(pbase + lmod);
#pragma unroll
        for (int j = 0; j < 8; ++j) {
            const int c = j + 8 * hi;
            if (c < NCEN) {
                float d2 = acc[j] + pnorm;
                uint64_t key = ((uint64_t)ford(d2) << 14) | p;
                if (key <= prefix[c])
                    atomicOr(&bits[p >> 5], 1u << (p & 31));
            }
        }
    }
    __syncthreads();

    // ---- Count block points; remaining = 9830 - nblock (always > 0: nblock <= 8190)
    uint32_t cnt = 0;
    for (int i = tid; i < NPTS / 32; i += NTHREADS) cnt += __popc(bits[i]);
    red[tid] = cnt;
    __syncthreads();
    for (int s = NTHREADS / 2; s > 0; s >>= 1) {
        if (tid < s) red[tid] += red[tid + s];
        __syncthreads();
    }
    if (tid == 0) {
        uint32_t nblock = (uint32_t)red[0];
        remaining_s = (nblock < NUM_MASKS) ? (NUM_MASKS - nblock) : 0u;
        prefix[0] = 0;
        kth[0] = remaining_s;
    }
    for (int i = tid; i < 256; i += NTHREADS) hist[i] = 0;
    __syncthreads();

    // ---- Phase 3: radix-select remaining-th smallest rand_mask among non-block ----
    const uint32_t rem = remaining_s;   // uniform across block
    if (rem > 0) {
        for (int pass = 0; pass < 6; ++pass) {
            const int dsh = 40 - 8 * pass;
            for (int p = tid; p < NPTS; p += NTHREADS) {
                const bool blocked = (bits[p >> 5] >> (p & 31)) & 1u;
                if (!blocked) {
                    uint64_t key = ((uint64_t)ford(rm[p]) << 14) | (uint32_t)p;
                    if ((key >> (dsh + 8)) == prefix[0])
                        atomicAdd(&hist[(uint32_t)(key >> dsh) & 0xFFu], 1u);
                }
            }
            __syncthreads();
            if (tid == 0) {
                uint32_t k = kth[0], cum = 0;
                int d = 0;
                for (; d < 256; ++d) {
                    uint32_t h = hist[d];
                    if (cum + h >= k) break;
                    cum += h;
                }
                if (d > 255) d = 255;
                prefix[0] = (prefix[0] << 8) | (uint32_t)d;
                kth[0] = k - cum;
            }
            __syncthreads();
            for (int i = tid; i < 256; i += NTHREADS) hist[i] = 0;
            __syncthreads();
        }
        if (tid == 0) thresh2 = prefix[0];
    } else {
        if (tid == 0) thresh2 = 0;   // ordered bits of rand>0 are >= 1<<45 -> never selected
    }
    __syncthreads();

    // ---- Phase 4: write final mask as float 0/1 ----
    float* o = out + (size_t)bm * NPTS;
    const uint64_t t2 = thresh2;
    for (int p = tid; p < NPTS; p += NTHREADS) {
        const bool blocked = (bits[p >> 5] >> (p & 31)) & 1u;
        bool sel2 = false;
        if (!blocked) {
            uint64_t key = ((uint64_t)ford(rm[p]) << 14) | (uint32_t)p;
            sel2 = (key <= t2);
        }
        o[p] = (blocked || sel2) ? 1.0f : 0.0f;
    }
}

extern "C" void kernel_launch(void* const* d_in, const int* in_sizes, int n_in,
                              void* d_out, int out_size, void* d_ws, size_t ws_size,
                              hipStream_t stream) {
    (void)in_sizes; (void)n_in; (void)d_ws; (void)ws_size; (void)out_size;
    const float* centers      = (const float*)d_in[0];  // [64, 16384, 3] f32
    const float* rand_centers = (const float*)d_in[1];  // [128, 16384] f32
    const float* rand_mask    = (const float*)d_in[2];  // [128, 16384] f32
    float* out = (float*)d_out;                         // [128, 16384] mask as 0/1 f32

    // CDNA5 allows up to 320KB LDS per workgroup; opt in for the 192KB dynamic slab.
    static_assert(COORD_BYTES == 196608, "coord staging size");
    (void)hipFuncSetAttribute((const void*)BlockPatchMasking_72241349919073_kernel,
                              hipFuncAttributeMaxDynamicSharedMemorySize, COORD_BYTES);

    hipLaunchKernelGGL(BlockPatchMasking_72241349919073_kernel,
                       dim3(BM), dim3(NTHREADS), COORD_BYTES, stream,
                       centers, rand_centers, rand_mask, out);
}